// PIT_SISNR_3272765079734
// MI455X (gfx1250) — compile-verified
//
#include <hip/hip_runtime.h>
#include <hip/hip_bf16.h>

// PIT SI-SNR loss, B=32, S=2, T=256000, f32.
// Pass 1: one-pass raw-moment reduction (memory-bound, ~131MB -> ~5.6us @ 23.3TB/s).
// Pass 2: one wave32 sums per-chunk partials with V_WMMA_F32_16X16X4_F32 (A = stats x chunks,
//         B = ones => exact f32 row sums), then does the SI-SNR / PIT scalar math.

#define T_LEN   256000
#define NCHUNK  40          // multiple of 4 (WMMA K=4 groups); (T/4)/NCHUNK = 1600 exactly
#define BLOCK   256
#define EPS_F   1.1920928955078125e-07f

typedef float v2f __attribute__((ext_vector_type(2)));
typedef float v8f __attribute__((ext_vector_type(8)));

// stat layout (rows M of the WMMA A matrix / padded partial records):
// 0:Sp0 1:Sp1 2:St0 3:St1 4:Qp0 5:Qp1 6:Qt0 7:Qt1 8:d00 9:d01 10:d10 11:d11 (12..15 zero pad)

__device__ __forceinline__ void acc1(float* acc, float pa, float pb, float ta, float tb) {
  acc[0] += pa; acc[1] += pb; acc[2] += ta; acc[3] += tb;
  acc[4]  = fmaf(pa, pa, acc[4]);
  acc[5]  = fmaf(pb, pb, acc[5]);
  acc[6]  = fmaf(ta, ta, acc[6]);
  acc[7]  = fmaf(tb, tb, acc[7]);
  acc[8]  = fmaf(pa, ta, acc[8]);
  acc[9]  = fmaf(pa, tb, acc[9]);
  acc[10] = fmaf(pb, ta, acc[10]);
  acc[11] = fmaf(pb, tb, acc[11]);
}

__global__ __launch_bounds__(BLOCK) void sisnr_stats(const float* __restrict__ preds,
                                                     const float* __restrict__ tgts,
                                                     float* __restrict__ ws) {
  const int b     = blockIdx.x / NCHUNK;
  const int chunk = blockIdx.x % NCHUNK;
  const int q4    = T_LEN / 4;       // float4 count per signal
  const int per   = q4 / NCHUNK;     // 1600 float4 per chunk per signal
  const int start = chunk * per;
  const int end   = start + per;

  const float4* p0 = (const float4*)(preds + (size_t)b * 2 * T_LEN);
  const float4* p1 = p0 + q4;
  const float4* t0 = (const float4*)(tgts + (size_t)b * 2 * T_LEN);
  const float4* t1 = t0 + q4;

  float acc[12];
#pragma unroll
  for (int k = 0; k < 12; ++k) acc[k] = 0.f;

  for (int i = start + (int)threadIdx.x; i < end; i += BLOCK) {
    __builtin_prefetch(&p0[i + BLOCK], 0, 0);   // global_prefetch_b8, speculative
    __builtin_prefetch(&t0[i + BLOCK], 0, 0);
    float4 a  = p0[i];
    float4 bq = p1[i];
    float4 c  = t0[i];
    float4 d  = t1[i];
    acc1(acc, a.x, bq.x, c.x, d.x);
    acc1(acc, a.y, bq.y, c.y, d.y);
    acc1(acc, a.z, bq.z, c.z, d.z);
    acc1(acc, a.w, bq.w, c.w, d.w);
  }

  // wave32 tree reduction (CDNA5 is wave32: width must be 32, not 64)
#pragma unroll
  for (int k = 0; k < 12; ++k) {
    float v = acc[k];
#pragma unroll
    for (int off = 16; off > 0; off >>= 1) v += __shfl_xor(v, off, 32);
    acc[k] = v;
  }

  __shared__ float red[BLOCK / 32][12];
  const int wave = threadIdx.x >> 5;
  const int lane = threadIdx.x & 31;
  if (lane == 0) {
#pragma unroll
    for (int k = 0; k < 12; ++k) red[wave][k] = acc[k];
  }
  __syncthreads();

  // one padded record of 16 floats per (batch,chunk); rows 12..15 zeroed for WMMA A padding
  if (threadIdx.x < 16) {
    float s = 0.f;
    if (threadIdx.x < 12) {
#pragma unroll
      for (int w = 0; w < BLOCK / 32; ++w) s += red[w][threadIdx.x];
    }
    ws[(size_t)blockIdx.x * 16 + threadIdx.x] = s;
  }
}

__global__ __launch_bounds__(32) void sisnr_finalize(const float* __restrict__ ws,
                                                     float* __restrict__ out, int B) {
  const int lane = threadIdx.x;          // exactly one wave32, EXEC all ones (uniform flow)
  const int m    = lane & 15;            // A row M = stat index
  const int koff = (lane >> 4) * 2;      // A layout: lanes 0-15 -> K=0,1 ; lanes 16-31 -> K=2,3
  v2f ones; ones.x = 1.0f; ones.y = 1.0f;   // B = 4x16 all-ones -> D[m][*] = sum_k A[m][k]
  const float invT = 1.0f / (float)T_LEN;

  float sum_best = 0.f;
  for (int b = 0; b < B; ++b) {
    const float* base = ws + (size_t)b * NCHUNK * 16;
    v8f c = {};
    for (int g = 0; g < NCHUNK; g += 4) {
      v2f a;
      a.x = base[(g + koff)     * 16 + m];
      a.y = base[(g + koff + 1) * 16 + m];
      // D = A(16x4 f32) * ones(4x16) + C : exact f32 chunk summation per stat row
      c = __builtin_amdgcn_wmma_f32_16x16x4_f32(false, a, false, ones,
                                                (short)0, c, false, false);
    }
    // C/D layout: VGPR r -> M=r (lanes 0-15), M=8+r (lanes 16-31); all N columns identical
    float carr[8];
#pragma unroll
    for (int r = 0; r < 8; ++r) carr[r] = c[r];
    float st[12];
#pragma unroll
    for (int k = 0; k < 12; ++k) st[k] = __shfl(carr[k & 7], (k < 8) ? 0 : 16, 32);

    // mean-corrected moments
    float ppv[2] = { st[4] - st[0] * st[0] * invT, st[5] - st[1] * st[1] * invT };
    float ttv[2] = { st[6] - st[2] * st[2] * invT, st[7] - st[3] * st[3] * invT };
    float dv[2][2] = { { st[8]  - st[0] * st[2] * invT, st[9]  - st[0] * st[3] * invT },
                       { st[10] - st[1] * st[2] * invT, st[11] - st[1] * st[3] * invT } };
    float sis[2][2];
#pragma unroll
    for (int i = 0; i < 2; ++i) {
#pragma unroll
      for (int j = 0; j < 2; ++j) {
        float alpha = dv[i][j] / (ttv[j] + EPS_F);
        float starg = alpha * alpha * ttv[j];
        float noise = ppv[i] - 2.0f * alpha * dv[i][j] + starg;
        sis[i][j] = 10.0f * log10f((starg + EPS_F) / (noise + EPS_F));
      }
    }
    float perm0 = 0.5f * (sis[0][0] + sis[1][1]);   // identity permutation
    float perm1 = 0.5f * (sis[0][1] + sis[1][0]);   // swapped
    sum_best += fmaxf(perm0, perm1);
  }
  if (lane == 0) out[0] = -sum_best / (float)B;
}

extern "C" void kernel_launch(void* const* d_in, const int* in_sizes, int n_in,
                              void* d_out, int out_size, void* d_ws, size_t ws_size,
                              hipStream_t stream) {
  const float* preds = (const float*)d_in[0];
  const float* tgts  = (const float*)d_in[1];
  float* out = (float*)d_out;
  float* ws  = (float*)d_ws;     // needs B*NCHUNK*16*4 = 80 KB
  const int B = in_sizes[0] / (2 * T_LEN);   // 32

  sisnr_stats<<<dim3(B * NCHUNK), dim3(BLOCK), 0, stream>>>(preds, tgts, ws);
  sisnr_finalize<<<dim3(1), dim3(32), 0, stream>>>(ws, out, B);

  (void)n_in; (void)out_size; (void)ws_size;
}